// Chord_decoder_33191507263659
// MI455X (gfx1250) — compile-verified
//
#include <hip/hip_runtime.h>
#include <hip/hip_bf16.h>

// H=1024, E=512, V=50257, L=4096
#define Hn 1024
#define En 512
#define Vn 50257
#define Ln 4096

typedef float v2f __attribute__((ext_vector_type(2)));
typedef float v8f __attribute__((ext_vector_type(8)));

// ---------------------------------------------------------------------------
// Async tile feed: each wave owns 16 rows of the 128-row block tile and copies
// them itself with global_load_async_to_lds_b128 (ASYNCcnt-tracked DMA into
// LDS, no VGPR round trip, no cross-wave barrier). 8 instructions move
// 16 rows x 64 floats (wave-wide 512B each).
// LDS row stride 68 dwords (68 mod 64 = 4) => WMMA A-operand ds reads are
// bank-conflict-free.
// ---------------------------------------------------------------------------
__device__ __forceinline__ void issue_tile(const float* __restrict__ W, int K,
                                           int r0w, int M, int kc,
                                           float* wbuf, int lane)
{
    #pragma unroll
    for (int i = 0; i < 8; ++i) {
        int idx = i * 32 + lane;    // 0..255 over the wave
        int rl  = idx >> 4;         // 0..15 local row
        int c4  = idx & 15;         // float4 column within 64-float chunk
        int gr  = r0w + rl; if (gr >= M) gr = M - 1;
        const float* src = W + (size_t)gr * K + kc + c4 * 4;
        unsigned lds = (unsigned)(size_t)(wbuf + rl * 68 + c4 * 4);
        asm volatile("global_load_async_to_lds_b128 %0, %1, off"
                     :: "v"(lds), "v"((unsigned long long)(size_t)src)
                     : "memory");
    }
}

// ---------------------------------------------------------------------------
// WMMA matvec accumulator, double-buffered async pipeline.
// Block = 256 threads (8 waves), block covers 128 rows (16 per wave).
// A tile (16x4 f32): lane<16 -> row=lane, elems {K,K+1}; lane>=16 -> row=lane-16,
// elems {K+2,K+3}. B = vector chunk broadcast across all 16 N columns.
// All D columns equal => lane0 c[0..7] = y[r0w..r0w+7], lane16 = y[r0w+8..+15].
// ---------------------------------------------------------------------------
__device__ __forceinline__ void wmma_mv(v8f& c, const float* __restrict__ W, int K,
                                        int r0, int M, const float* __restrict__ xlds,
                                        float* tile0, float* tile1)
{
    const int t    = threadIdx.x;
    const int wv   = t >> 5;
    const int lane = t & 31;
    const int lr   = lane & 15;
    const int hi   = lane >> 4;
    float* wbuf0 = tile0 + wv * 16 * 68;
    float* wbuf1 = tile1 + wv * 16 * 68;
    const int r0w = r0 + wv * 16;
    const int nch = K / 64;

    issue_tile(W, K, r0w, M, 0, wbuf0, lane);
    for (int ch = 0; ch < nch; ++ch) {
        float* cur = (ch & 1) ? wbuf1 : wbuf0;
        float* nxt = (ch & 1) ? wbuf0 : wbuf1;
        if (ch + 1 < nch) {
            issue_tile(W, K, r0w, M, (ch + 1) * 64, nxt, lane);
            asm volatile("s_wait_asynccnt 8" ::: "memory");   // chunk ch landed
        } else {
            asm volatile("s_wait_asynccnt 0" ::: "memory");
        }
        const float* ar = cur + lr * 68 + 2 * hi;
        const int kb = ch * 64;
        #pragma unroll
        for (int k = 0; k < 64; k += 4) {
            v2f a = *(const v2f*)(ar + k);
            v2f b;
            b.x = xlds[kb + k + 2 * hi];
            b.y = xlds[kb + k + 2 * hi + 1];
            c = __builtin_amdgcn_wmma_f32_16x16x4_f32(
                    false, a, false, b, (short)0, c, false, false);
        }
    }
}

// ---------------------------------------------------------------------------
// Phase 1: scores[l] = enc[l,:] . h   (one wave per row, coalesced float4)
// ---------------------------------------------------------------------------
__global__ void scores_kernel(const float* __restrict__ enc, const float* __restrict__ h,
                              float* __restrict__ scores)
{
    int t = threadIdx.x, wv = t >> 5, lane = t & 31;
    int row = blockIdx.x * 8 + wv;
    const float4* rp = (const float4*)(enc + (size_t)row * Hn);
    const float4* hp = (const float4*)h;
    float sum = 0.f;
    #pragma unroll
    for (int i = lane; i < Hn / 4; i += 32) {
        float4 a = rp[i], b = hp[i];
        sum += a.x * b.x + a.y * b.y + a.z * b.z + a.w * b.w;
    }
    #pragma unroll
    for (int m = 16; m >= 1; m >>= 1) sum += __shfl_xor(sum, m, 32);
    if (lane == 0) scores[row] = sum;
}

// ---------------------------------------------------------------------------
// Phase 2: softmax over L=4096 in-place; also zero the context accumulator.
// ---------------------------------------------------------------------------
__global__ void softmax_kernel(float* __restrict__ scores, float* __restrict__ context)
{
    __shared__ float red[1024];
    int t = threadIdx.x;
    float v[4];
    float m = -3.0e38f;
    #pragma unroll
    for (int i = 0; i < 4; ++i) { v[i] = scores[t + i * 1024]; m = fmaxf(m, v[i]); }
    red[t] = m; __syncthreads();
    for (int s = 512; s > 0; s >>= 1) { if (t < s) red[t] = fmaxf(red[t], red[t + s]); __syncthreads(); }
    m = red[0]; __syncthreads();
    float sum = 0.f;
    #pragma unroll
    for (int i = 0; i < 4; ++i) { v[i] = expf(v[i] - m); sum += v[i]; }
    red[t] = sum; __syncthreads();
    for (int s = 512; s > 0; s >>= 1) { if (t < s) red[t] += red[t + s]; __syncthreads(); }
    float inv = 1.0f / red[0];
    #pragma unroll
    for (int i = 0; i < 4; ++i) scores[t + i * 1024] = v[i] * inv;
    context[t] = 0.0f;
}

// ---------------------------------------------------------------------------
// Phase 3: context[h] += sum_l w[l] * enc[l,h]   (coalesced across h, f32 atomics)
// ---------------------------------------------------------------------------
__global__ void context_kernel(const float* __restrict__ w, const float* __restrict__ enc,
                               float* __restrict__ context)
{
    int t  = threadIdx.x;            // 0..1023 -> h index
    int l0 = blockIdx.x * 256;       // 16 blocks
    float sum = 0.f;
    for (int l = l0; l < l0 + 256; ++l)
        sum = fmaf(w[l], enc[(size_t)l * Hn + t], sum);
    unsafeAtomicAdd(&context[t], sum);
}

// ---------------------------------------------------------------------------
// Phase 4: x[0:512] = attn_W @ context + attn_b ; x[512:1024] = emb[token]
// ---------------------------------------------------------------------------
__global__ void attnx_kernel(const float* __restrict__ context, const float* __restrict__ attnW,
                             const float* __restrict__ attnb, const float* __restrict__ emb,
                             const int* __restrict__ token, float* __restrict__ x)
{
    if (blockIdx.x >= 64) {
        int i = (blockIdx.x - 64) * 256 + threadIdx.x;   // 0..511
        x[En + i] = emb[(size_t)token[0] * En + i];
        return;
    }
    int t = threadIdx.x, wv = t >> 5, lane = t & 31;
    int row = blockIdx.x * 8 + wv;                       // 0..511
    const float4* rp = (const float4*)(attnW + (size_t)row * Hn);
    const float4* cp = (const float4*)context;
    float sum = 0.f;
    #pragma unroll
    for (int i = lane; i < Hn / 4; i += 32) {
        float4 a = rp[i], b = cp[i];
        sum += a.x * b.x + a.y * b.y + a.z * b.z + a.w * b.w;
    }
    #pragma unroll
    for (int m = 16; m >= 1; m >>= 1) sum += __shfl_xor(sum, m, 32);
    if (lane == 0) x[row] = sum + attnb[row];
}

// ---------------------------------------------------------------------------
// Phase 5: gates = W_ih @ x + b_ih + W_hh @ h + b_hh    (WMMA path, 32 MB)
// ---------------------------------------------------------------------------
__global__ void gates_kernel(const float* __restrict__ Wih, const float* __restrict__ Whh,
                             const float* __restrict__ bih, const float* __restrict__ bhh,
                             const float* __restrict__ x, const float* __restrict__ h,
                             float* __restrict__ gates)
{
    __shared__ __align__(16) float tile0[128 * 68];
    __shared__ __align__(16) float tile1[128 * 68];
    __shared__ float xs[Hn], hs[Hn];
    for (int i = threadIdx.x; i < Hn; i += 256) { xs[i] = x[i]; hs[i] = h[i]; }
    __syncthreads();

    v8f c = {};
    int r0 = blockIdx.x * 128;
    wmma_mv(c, Wih, Hn, r0, 4 * Hn, xs, tile0, tile1);
    wmma_mv(c, Whh, Hn, r0, 4 * Hn, hs, tile0, tile1);

    int t = threadIdx.x, wv = t >> 5, lane = t & 31;
    int rb = r0 + wv * 16;
    if (lane == 0) {
        #pragma unroll
        for (int i = 0; i < 8; ++i) { int r = rb + i;     gates[r] = c[i] + bih[r] + bhh[r]; }
    } else if (lane == 16) {
        #pragma unroll
        for (int i = 0; i < 8; ++i) { int r = rb + 8 + i; gates[r] = c[i] + bih[r] + bhh[r]; }
    }
}

// ---------------------------------------------------------------------------
// Phase 6: LSTM pointwise; writes h_new (ws + d_out) and c_new (d_out)
// ---------------------------------------------------------------------------
__global__ void lstm_kernel(const float* __restrict__ gates, const float* __restrict__ c_old,
                            float* __restrict__ hnew_ws, float* __restrict__ out)
{
    int t = threadIdx.x;   // 0..1023
    float ig = 1.f / (1.f + expf(-gates[t]));
    float fg = 1.f / (1.f + expf(-gates[t + Hn]));
    float gg = tanhf(gates[t + 2 * Hn]);
    float og = 1.f / (1.f + expf(-gates[t + 3 * Hn]));
    float cn = fg * c_old[t] + ig * gg;
    float hn = og * tanhf(cn);
    hnew_ws[t]          = hn;
    out[Vn + t]         = hn;   // h_new output
    out[Vn + Hn + t]    = cn;   // c_new output
}

// ---------------------------------------------------------------------------
// Phase 7: logits = out_W @ h_new + out_b    (WMMA path, 206 MB, dominant)
// ---------------------------------------------------------------------------
__global__ void logits_kernel(const float* __restrict__ outW, const float* __restrict__ outb,
                              const float* __restrict__ hnew, float* __restrict__ out)
{
    __shared__ __align__(16) float tile0[128 * 68];
    __shared__ __align__(16) float tile1[128 * 68];
    __shared__ float hs[Hn];
    for (int i = threadIdx.x; i < Hn; i += 256) hs[i] = hnew[i];
    __syncthreads();

    v8f c = {};
    int r0 = blockIdx.x * 128;
    wmma_mv(c, outW, Hn, r0, Vn, hs, tile0, tile1);

    int t = threadIdx.x, wv = t >> 5, lane = t & 31;
    int rb = r0 + wv * 16;
    if (lane == 0) {
        #pragma unroll
        for (int i = 0; i < 8; ++i) { int r = rb + i;     if (r < Vn) out[r] = c[i] + outb[r]; }
    } else if (lane == 16) {
        #pragma unroll
        for (int i = 0; i < 8; ++i) { int r = rb + 8 + i; if (r < Vn) out[r] = c[i] + outb[r]; }
    }
}

// ---------------------------------------------------------------------------
extern "C" void kernel_launch(void* const* d_in, const int* in_sizes, int n_in,
                              void* d_out, int out_size, void* d_ws, size_t ws_size,
                              hipStream_t stream)
{
    (void)in_sizes; (void)n_in; (void)out_size; (void)ws_size;
    const int*   token = (const int*)  d_in[0];
    const float* hn    = (const float*)d_in[1];
    const float* cn    = (const float*)d_in[2];
    const float* enc   = (const float*)d_in[3];
    const float* emb   = (const float*)d_in[4];
    const float* Wih   = (const float*)d_in[5];
    const float* Whh   = (const float*)d_in[6];
    const float* bih   = (const float*)d_in[7];
    const float* bhh   = (const float*)d_in[8];
    const float* attnW = (const float*)d_in[9];
    const float* attnb = (const float*)d_in[10];
    const float* outW  = (const float*)d_in[11];
    const float* outb  = (const float*)d_in[12];

    float* out = (float*)d_out;         // [logits (50257) | h_new (1024) | c_new (1024)]
    float* ws  = (float*)d_ws;
    float* scores  = ws;                // 4096 (becomes softmax weights in place)
    float* context = ws + 4096;         // 1024
    float* x       = ws + 5120;         // 1024
    float* gates   = ws + 6144;         // 4096
    float* hnew    = ws + 10240;        // 1024

    scores_kernel <<<Ln / 8, 256, 0, stream>>>(enc, hn, scores);
    softmax_kernel<<<1, 1024, 0, stream>>>(scores, context);
    context_kernel<<<Ln / 256, 1024, 0, stream>>>(scores, enc, context);
    attnx_kernel  <<<66, 256, 0, stream>>>(context, attnW, attnb, emb, token, x);
    gates_kernel  <<<(4 * Hn) / 128, 256, 0, stream>>>(Wih, Whh, bih, bhh, x, hn, gates);
    lstm_kernel   <<<1, 1024, 0, stream>>>(gates, cn, hnew, out);
    logits_kernel <<<(Vn + 127) / 128, 256, 0, stream>>>(outW, outb, hnew, out);
}